// ConvPolicy_85169201480379
// MI455X (gfx1250) — compile-verified
//
#include <hip/hip_runtime.h>

typedef __attribute__((ext_vector_type(2))) float v2f;
typedef __attribute__((ext_vector_type(4))) float f4v;
typedef __attribute__((ext_vector_type(8))) float v8f;

#define RPB 256  // rows per block == threads per block (8 waves of 32)

__device__ __forceinline__ float relu_(float v) { return v > 0.0f ? v : 0.0f; }

__global__ __launch_bounds__(RPB) void ConvPolicy_kernel(
    const float* __restrict__ x,
    const float* __restrict__ fc_obs_w, const float* __restrict__ fc_obs_b,
    const float* __restrict__ fc_emb_w, const float* __restrict__ fc_emb_b,
    const float* __restrict__ conv1_w,  const float* __restrict__ conv1_b,
    const float* __restrict__ conv2_w,  const float* __restrict__ conv2_b,
    const float* __restrict__ deconv1_w, const float* __restrict__ deconv1_b,
    const float* __restrict__ deconv2_w, const float* __restrict__ deconv2_b,
    float* __restrict__ out, int nrows)
{
    __shared__ float xs[RPB * 18];        // staged input rows   (18 KB)
    __shared__ float os[RPB * 7];         // staged output rows  ( 7 KB)
    __shared__ float fcobs[8][32][2];     // WMMA fc_obs results ( 2 KB)

    const int tid  = threadIdx.x;
    const int base = blockIdx.x * RPB;
    const int rows = (nrows - base < RPB) ? (nrows - base) : RPB;

    // ---- stage 256 rows (18 dwords each) into LDS, fully coalesced B128 ----
    if (rows == RPB) {
        const f4v* gx4 = (const f4v*)(x + (size_t)base * 18); // base*18 % 4 == 0
        f4v* xs4 = (f4v*)xs;
        for (int k = tid; k < RPB * 18 / 4; k += RPB) xs4[k] = gx4[k];
    } else {
        const int nd = rows * 18;
        for (int k = tid; k < nd; k += RPB) xs[k] = x[(size_t)base * 18 + k];
    }
    __syncthreads();

    // ---- uniform tiny weights (scalar loads) ----
    const float ob0 = fc_obs_b[0], ob1 = fc_obs_b[1];
    const float w1a0 = conv1_w[0], w1a1 = conv1_w[1], w1a2 = conv1_w[2];
    const float w1b0 = conv1_w[3], w1b1 = conv1_w[4], w1b2 = conv1_w[5];
    const float cb1 = conv1_b[0];
    const float w20 = conv2_w[0], w21 = conv2_w[1];
    const float cb2 = conv2_b[0];
    const float ew00 = fc_emb_w[0], ew01 = fc_emb_w[1];
    const float ew10 = fc_emb_w[2], ew11 = fc_emb_w[3];
    const float eb0 = fc_emb_b[0], eb1 = fc_emb_b[1];
    const float d1w0 = deconv1_w[0], d1w1 = deconv1_w[1], d1b = deconv1_b[0];
    const float d2w0 = deconv2_w[0], d2w1 = deconv2_w[1], d2w2 = deconv2_w[2];
    const float d2b = deconv2_b[0];

    // ---- fc_obs via V_WMMA_F32_16X16X4_F32 (exact fp32, K=4 == obs width) ----
    const int lane = tid & 31;
    const int wv   = tid >> 5;
    const int n16  = lane & 15;   // N index for B/D; M index for A
    const int hi   = lane >> 4;   // lane half
    const int k0   = hi * 2;      // K pair held by this lane (ISA 16x4 f32 A layout)

    // B = fc_obs_w^T zero-padded to 4x16: B(k,n) = fc_obs_w[n*4+k] for n<2.
    // Assumed 4x16 f32 B layout (mirror of C/D striping): VGPR0 -> K = 0 / 2
    // for lane halves, VGPR1 -> K = 1 / 3.
    v2f bmat;
    bmat.x = (n16 < 2) ? fc_obs_w[n16 * 4 + k0]     : 0.0f;
    bmat.y = (n16 < 2) ? fc_obs_w[n16 * 4 + k0 + 1] : 0.0f;

    for (int half = 0; half < 2; ++half) {
        const int mrow = wv * 32 + half * 16 + n16;  // row-in-block for A row M=n16
        v2f amat;
        amat.x = xs[mrow * 18 + k0];       // obs[k0]
        amat.y = xs[mrow * 18 + k0 + 1];   // obs[k0+1]
        v8f cacc = {};
        v8f d = __builtin_amdgcn_wmma_f32_16x16x4_f32(
            false, amat, false, bmat, (short)0, cacc, false, false);
        if (n16 < 2) {                     // only columns 0,1 are meaningful
#pragma unroll
            for (int r = 0; r < 8; ++r)    // D: VGPR r -> M = r + 8*hi
                fcobs[wv][half * 16 + r + hi * 8][n16] = d[r];
        }
    }
    __syncthreads();

    // ---- scalar per-row pipeline: thread tid owns row base+tid ----
    {
        const float* xr = xs + tid * 18;
        const float f0 = relu_(fcobs[wv][lane][0] + ob0);
        const float f1 = relu_(fcobs[wv][lane][1] + ob1);

        // conv1: 2-in-ch, k=3, stride 2  (j = xr[4..10], jd = xr[11..17])
        float c1[3];
#pragma unroll
        for (int i = 0; i < 3; ++i) {
            const float s = cb1
                + w1a0 * xr[4 + 2 * i] + w1a1 * xr[5 + 2 * i] + w1a2 * xr[6 + 2 * i]
                + w1b0 * xr[11 + 2 * i] + w1b1 * xr[12 + 2 * i] + w1b2 * xr[13 + 2 * i];
            c1[i] = relu_(s);
        }
        // conv2: k=2, stride 1
        const float c20 = relu_(cb2 + w20 * c1[0] + w21 * c1[1]);
        const float c21 = relu_(cb2 + w20 * c1[1] + w21 * c1[2]);

        // fc_emb on (fc_obs + fm)
        const float h0 = f0 + c20, h1 = f1 + c21;
        const float e0 = relu_(eb0 + ew00 * h0 + ew01 * h1);
        const float e1 = relu_(eb1 + ew10 * h0 + ew11 * h1);

        // deconv1: transpose conv, k=2, s=1: out[n] = sum_{i+t=n} e[i]*w[t]
        const float t0 = relu_(d1b + e0 * d1w0);
        const float t1 = relu_(d1b + e0 * d1w1 + e1 * d1w0);
        const float t2 = relu_(d1b + e1 * d1w1);

        // deconv2: transpose conv, k=3, s=2: out[n] = sum_{2i+t=n} t[i]*w[t]
        float* o = os + tid * 7;
        o[0] = d2b + t0 * d2w0;
        o[1] = d2b + t0 * d2w1;
        o[2] = d2b + t0 * d2w2 + t1 * d2w0;
        o[3] = d2b + t1 * d2w1;
        o[4] = d2b + t1 * d2w2 + t2 * d2w0;
        o[5] = d2b + t2 * d2w1;
        o[6] = d2b + t2 * d2w2;
    }
    __syncthreads();

    // ---- coalesced B128 store of staged outputs ----
    if (rows == RPB) {
        f4v* go4 = (f4v*)(out + (size_t)base * 7);   // base*7 % 4 == 0
        const f4v* os4 = (const f4v*)os;
        for (int k = tid; k < RPB * 7 / 4; k += RPB) go4[k] = os4[k];
    } else {
        const int nd = rows * 7;
        for (int k = tid; k < nd; k += RPB) out[(size_t)base * 7 + k] = os[k];
    }
}

extern "C" void kernel_launch(void* const* d_in, const int* in_sizes, int n_in,
                              void* d_out, int out_size, void* d_ws, size_t ws_size,
                              hipStream_t stream) {
    const float* x         = (const float*)d_in[0];
    const float* fc_obs_w  = (const float*)d_in[1];
    const float* fc_obs_b  = (const float*)d_in[2];
    const float* fc_emb_w  = (const float*)d_in[3];
    const float* fc_emb_b  = (const float*)d_in[4];
    const float* conv1_w   = (const float*)d_in[5];
    const float* conv1_b   = (const float*)d_in[6];
    const float* conv2_w   = (const float*)d_in[7];
    const float* conv2_b   = (const float*)d_in[8];
    const float* deconv1_w = (const float*)d_in[9];
    const float* deconv1_b = (const float*)d_in[10];
    const float* deconv2_w = (const float*)d_in[11];
    const float* deconv2_b = (const float*)d_in[12];

    const int nrows = in_sizes[0] / 18;
    const int nblocks = (nrows + RPB - 1) / RPB;

    ConvPolicy_kernel<<<nblocks, RPB, 0, stream>>>(
        x, fc_obs_w, fc_obs_b, fc_emb_w, fc_emb_b,
        conv1_w, conv1_b, conv2_w, conv2_b,
        deconv1_w, deconv1_b, deconv2_w, deconv2_b,
        (float*)d_out, nrows);
}